// GraphEncoder_9929964388988
// MI455X (gfx1250) — compile-verified
//
#include <hip/hip_runtime.h>

#define D 64
#define NEG_SLOPE 0.2f

typedef __attribute__((ext_vector_type(2))) float v2f;
typedef __attribute__((ext_vector_type(8))) float v8f;

// Monotone float<->int encoding so float max can use integer atomicMax.
__device__ __forceinline__ int fenc(float f) {
    int i = __float_as_int(f);
    return i ^ ((i >> 31) & 0x7FFFFFFF);
}
__device__ __forceinline__ float fdec(int i) {
    return __int_as_float(i ^ ((i >> 31) & 0x7FFFFFFF));
}

// h = x @ W : (N x 64) * (64 x 64), fp32 WMMA 16x16x4.
// One 16x16 C tile per wave; 8 waves/block -> 32 rows x 64 cols per block.
__global__ void gemm_xw(const float* __restrict__ x, const float* __restrict__ W,
                        float* __restrict__ h, int N) {
    const int lane = threadIdx.x & 31;
    const int wave = threadIdx.x >> 5;             // 0..7
    const int m0 = blockIdx.x * 32 + (wave >> 2) * 16;
    const int n0 = (wave & 3) * 16;
    const int r  = lane & 15;
    const int ka = (lane < 16) ? 0 : 2;            // K split: lower lanes K+0/1, upper K+2/3
    int row = m0 + r; if (row >= N) row = N - 1;   // clamp loads; stores are guarded

    v8f c = {};
    #pragma unroll
    for (int k0 = 0; k0 < D; k0 += 4) {
        v2f a, b;
        a.x = x[row * D + k0 + ka];                // A 16x4: VGPR0=K{0,2}, VGPR1=K{1,3}
        a.y = x[row * D + k0 + ka + 1];
        b.x = W[(k0 + ka) * D + n0 + r];           // B 4x16: rows striped across lanes
        b.y = W[(k0 + ka + 1) * D + n0 + r];
        c = __builtin_amdgcn_wmma_f32_16x16x4_f32(
                false, a, false, b, (short)0, c, false, false);
    }
    #pragma unroll
    for (int v = 0; v < 8; ++v) {                  // C: VGPR v -> M = v (+8 for upper lanes)
        int orow = m0 + v + ((lane < 16) ? 0 : 8);
        if (orow < N) h[orow * D + n0 + r] = c[v];
    }
}

// Per-node attention logits: al_src[n] = h[n,:]·a_src, al_dst[n] = h[n,:]·a_dst.
// One wave per node (wave32: two 32-wide strips of the 64-dim row).
__global__ void node_logits(const float* __restrict__ h,
                            const float* __restrict__ a_src,
                            const float* __restrict__ a_dst,
                            float* __restrict__ al_src, float* __restrict__ al_dst,
                            int N) {
    int node = blockIdx.x * 8 + (threadIdx.x >> 5);
    int lane = threadIdx.x & 31;
    if (node >= N) return;
    float h0 = h[node * D + lane];
    float h1 = h[node * D + 32 + lane];
    float s = h0 * a_src[lane] + h1 * a_src[32 + lane];
    float d = h0 * a_dst[lane] + h1 * a_dst[32 + lane];
    #pragma unroll
    for (int off = 16; off > 0; off >>= 1) {
        s += __shfl_down(s, off);
        d += __shfl_down(d, off);
    }
    if (lane == 0) { al_src[node] = s; al_dst[node] = d; }
}

// Reset accumulators each call (harness does not re-poison between replays).
__global__ void init_layer(float* __restrict__ acc, int* __restrict__ m_int,
                           float* __restrict__ denom, int N) {
    int i = blockIdx.x * blockDim.x + threadIdx.x;
    if (i < N * D) acc[i] = 0.0f;
    if (i < N) { m_int[i] = fenc(-3.0e38f); denom[i] = 0.0f; }
}

// Pass 1: e = leaky_relu(al_src[src]+al_dst[dst]); segment max over dst.
__global__ void edge_logits_max(const int* __restrict__ ei, int E, int Etot,
                                const float* __restrict__ al_src,
                                const float* __restrict__ al_dst,
                                float* __restrict__ e, int* __restrict__ m_int) {
    int i = blockIdx.x * blockDim.x + threadIdx.x;
    if (i >= Etot) return;
    int s = (i < E) ? ei[i]     : (i - E);   // self-loops appended past E
    int t = (i < E) ? ei[E + i] : (i - E);
    float ev = al_src[s] + al_dst[t];
    ev = (ev > 0.0f) ? ev : ev * NEG_SLOPE;
    e[i] = ev;
    atomicMax(&m_int[t], fenc(ev));
}

// Pass 2: e <- exp(e - m[dst]); denom[dst] += e.
__global__ void edge_exp_denom(const int* __restrict__ ei, int E, int Etot,
                               float* __restrict__ e, const int* __restrict__ m_int,
                               float* __restrict__ denom) {
    int i = blockIdx.x * blockDim.x + threadIdx.x;
    if (i >= Etot) return;
    int t = (i < E) ? ei[E + i] : (i - E);
    float ex = expf(e[i] - fdec(m_int[t]));
    e[i] = ex;
    atomicAdd(&denom[t], ex);
}

// Pass 3: acc[dst,:] += (e/denom[dst]) * h[src,:]. 64 threads per edge,
// fully coalesced 256B gather + 256B atomic scatter.
__global__ void edge_scatter(const int* __restrict__ ei, int E, int Etot,
                             const float* __restrict__ e, const float* __restrict__ denom,
                             const float* __restrict__ h, float* __restrict__ acc) {
    unsigned int gid = blockIdx.x * blockDim.x + threadIdx.x;
    int edge = (int)(gid >> 6);
    int col  = (int)(gid & 63u);
    if (edge >= Etot) return;
    int s = (edge < E) ? ei[edge]     : (edge - E);
    int t = (edge < E) ? ei[E + edge] : (edge - E);
    float alpha = e[edge] / denom[t];
    atomicAdd(&acc[t * D + col], alpha * h[s * D + col]);
}

// out = relu(acc + b)
__global__ void bias_relu(const float* __restrict__ acc, const float* __restrict__ b,
                          float* __restrict__ out, int n) {
    int i = blockIdx.x * blockDim.x + threadIdx.x;
    if (i >= n) return;
    float v = acc[i] + b[i & (D - 1)];
    out[i] = (v > 0.0f) ? v : 0.0f;
}

extern "C" void kernel_launch(void* const* d_in, const int* in_sizes, int n_in,
                              void* d_out, int out_size, void* d_ws, size_t ws_size,
                              hipStream_t stream) {
    const float* x      = (const float*)d_in[0];
    const int*   ei     = (const int*)d_in[1];
    const float* W1     = (const float*)d_in[2];
    const float* a_src1 = (const float*)d_in[3];
    const float* a_dst1 = (const float*)d_in[4];
    const float* b1     = (const float*)d_in[5];
    const float* W2     = (const float*)d_in[6];
    const float* a_src2 = (const float*)d_in[7];
    const float* a_dst2 = (const float*)d_in[8];
    const float* b2     = (const float*)d_in[9];
    float* out = (float*)d_out;

    const int N    = in_sizes[0] / D;
    const int E    = in_sizes[1] / 2;
    const int Etot = E + N;

    // Workspace layout (floats): hbuf[N*D] | acc[N*D] | al_s[N] | al_d[N] | m_int[N] | denom[N] | ebuf[Etot]
    float* ws    = (float*)d_ws;
    float* hbuf  = ws;
    float* acc   = hbuf + (size_t)N * D;
    float* al_s  = acc + (size_t)N * D;
    float* al_d  = al_s + N;
    int*   m_int = (int*)(al_d + N);
    float* denom = (float*)(m_int + N);
    float* ebuf  = denom + N;

    const int TB = 256;
    dim3 blk(TB);
    unsigned g_gemm  = (N + 31) / 32;
    unsigned g_node  = (N + 7) / 8;
    unsigned g_nd    = ((unsigned)N * D + TB - 1) / TB;
    unsigned g_edge  = ((unsigned)Etot + TB - 1) / TB;
    unsigned g_scat  = ((unsigned)Etot * 64u + TB - 1) / TB;

    // ---------------- Layer 1: acc = relu(GAT(x; W1)) ----------------
    gemm_xw<<<g_gemm, blk, 0, stream>>>(x, W1, hbuf, N);
    node_logits<<<g_node, blk, 0, stream>>>(hbuf, a_src1, a_dst1, al_s, al_d, N);
    init_layer<<<g_nd, blk, 0, stream>>>(acc, m_int, denom, N);
    edge_logits_max<<<g_edge, blk, 0, stream>>>(ei, E, Etot, al_s, al_d, ebuf, m_int);
    edge_exp_denom<<<g_edge, blk, 0, stream>>>(ei, E, Etot, ebuf, m_int, denom);
    edge_scatter<<<g_scat, blk, 0, stream>>>(ei, E, Etot, ebuf, denom, hbuf, acc);
    bias_relu<<<g_nd, blk, 0, stream>>>(acc, b1, acc, N * D);

    // ---------------- Layer 2: out = relu(GAT(acc; W2)) ----------------
    gemm_xw<<<g_gemm, blk, 0, stream>>>(acc, W2, hbuf, N);
    node_logits<<<g_node, blk, 0, stream>>>(hbuf, a_src2, a_dst2, al_s, al_d, N);
    init_layer<<<g_nd, blk, 0, stream>>>(out, m_int, denom, N);
    edge_logits_max<<<g_edge, blk, 0, stream>>>(ei, E, Etot, al_s, al_d, ebuf, m_int);
    edge_exp_denom<<<g_edge, blk, 0, stream>>>(ei, E, Etot, ebuf, m_int, denom);
    edge_scatter<<<g_scat, blk, 0, stream>>>(ei, E, Etot, ebuf, denom, hbuf, out);
    bias_relu<<<g_nd, blk, 0, stream>>>(out, b2, out, N * D);
}